// EdgePromptplus_12824772346583
// MI455X (gfx1250) — compile-verified
//
#include <hip/hip_runtime.h>

typedef __attribute__((ext_vector_type(2))) float v2f;
typedef __attribute__((ext_vector_type(4))) float v4f;
typedef __attribute__((ext_vector_type(8))) float v8f;
typedef __attribute__((ext_vector_type(4))) int   v4i;

#define DIM 128
#define NA 32
#define TPB 512
#define WAVES 16
#define NEG_SLOPE 0.01f

// padded LDS strides (floats). All even (8B align), != 0 mod 64 banks.
#define COMB_STRIDE 132   // 16 rows (128 + pad4), rows 16B-aligned, stride%64 = 4
#define W_STRIDE    258   // W rows (256 + pad2)
#define AT_STRIDE   34    // anchor^T rows (32 + pad2)
#define PR_STRIDE   34    // probs rows (32 + pad2)

#define W_ELEMS      (NA * W_STRIDE)            // 8256
#define AT_ELEMS     (DIM * AT_STRIDE)          // 4352
#define COMB_ELEMS_W (16 * COMB_STRIDE)         // per-wave: 2112
#define PR_ELEMS_W   (16 * PR_STRIDE)           // per-wave: 544
#define SMEM_FLOATS  (W_ELEMS + AT_ELEMS + WAVES * (COMB_ELEMS_W + PR_ELEMS_W))

#if defined(__gfx1250__) && __has_builtin(__builtin_amdgcn_global_load_async_to_lds_b128)
#define USE_ASYNC_LDS 1
#else
#define USE_ASYNC_LDS 0
#endif

#if USE_ASYNC_LDS
typedef __attribute__((address_space(1))) v4i* gv4i_p;
typedef __attribute__((address_space(3))) v4i* lv4i_p;

__device__ __forceinline__ void wait_async0() {
    asm volatile("s_wait_asynccnt 0x0" ::: "memory");
}
__device__ __forceinline__ void wait_ds0() {
    asm volatile("s_wait_dscnt 0x0" ::: "memory");
}
__device__ __forceinline__ void async_row_to_lds(const float* gp, float* lp) {
    // per-lane 16B global -> LDS, tracked by ASYNCcnt
    __builtin_amdgcn_global_load_async_to_lds_b128(
        (gv4i_p)(uintptr_t)gp,
        (lv4i_p)(unsigned)(uintptr_t)lp,
        0, 0);
}
#endif

__global__ void edgeprompt_wmma_kernel(const float* __restrict__ x,
                                       const int* __restrict__ edge_index,
                                       const float* __restrict__ W,
                                       const float* __restrict__ b_lin,
                                       const float* __restrict__ anchor,
                                       float* __restrict__ out,
                                       int E) {
    extern __shared__ float smem[];
    float* Wlds = smem;                       // [32][258]
    float* aT   = Wlds + W_ELEMS;             // [128][34] : aT[n][k] = anchor[k][n]
    float* combBase = aT + AT_ELEMS;          // per-wave [16][132]
    float* prbBase  = combBase + WAVES * COMB_ELEMS_W;  // per-wave [16][34]

    const int tid  = threadIdx.x;
    const int lane = tid & 31;
    const int wave = tid >> 5;
    const int m    = lane & 15;        // M / N index within 16
    const int hs   = lane >> 4;        // half-select (lanes 16-31 hold K+2)
    const int bk   = hs * 2;           // K sub-offset within a 4-step

    // ---- stage W and anchor^T into LDS (whole block) ----
    for (int idx = tid; idx < NA * (2 * DIM); idx += TPB) {
        int r = idx >> 8, c = idx & 255;
        Wlds[r * W_STRIDE + c] = W[idx];
    }
    for (int idx = tid; idx < NA * DIM; idx += TPB) {
        int k = idx >> 7, n = idx & 127;
        aT[n * AT_STRIDE + k] = anchor[idx];
    }
    __syncthreads();

    float* comb = combBase + wave * COMB_ELEMS_W;
    float* prb  = prbBase  + wave * PR_ELEMS_W;

    const float bl0 = b_lin[m];
    const float bl1 = b_lin[16 + m];

    const int nTiles = (E + 15) >> 4;
    const int waveGlobal = blockIdx.x * WAVES + wave;
    const int waveStride = gridDim.x * WAVES;

    for (int tile = waveGlobal; tile < nTiles; tile += waveStride) {
        const int e0 = tile << 4;
        int eMine = e0 + m; if (eMine > E - 1) eMine = E - 1;
        const int sIdx = edge_index[eMine];
        const int dIdx = edge_index[E + eMine];

        v8f acc0 = {};  // anchors 0-15
        v8f acc1 = {};  // anchors 16-31

        // ===== logits GEMM, src half (K = 0..127) then dst half (K = 128..255) =====
        #pragma unroll
        for (int half = 0; half < 2; ++half) {
            // gather 16 rows of x into the per-wave staging tile
#if USE_ASYNC_LDS
            wait_ds0();   // all prior LDS reads/writes of comb complete before overwrite
            #pragma unroll 4
            for (int i = 0; i < 16; ++i) {
                int r = __shfl(half == 0 ? sIdx : dIdx, i, 32);
                async_row_to_lds(x + (size_t)r * DIM + lane * 4,
                                 comb + i * COMB_STRIDE + lane * 4);
            }
            wait_async0();  // async LDS writes visible before DS fragment reads
#else
            #pragma unroll 4
            for (int i = 0; i < 16; ++i) {
                int r = __shfl(half == 0 ? sIdx : dIdx, i, 32);
                v4f val = *(const v4f*)(x + (size_t)r * DIM + lane * 4);
                *(v4f*)(comb + i * COMB_STRIDE + lane * 4) = val;
            }
#endif
            const int wOff = half * DIM;  // W column offset for this half
            #pragma unroll 8
            for (int kk = 0; kk < 32; ++kk) {
                v2f a  = *(const v2f*)(comb + m * COMB_STRIDE + kk * 4 + bk);
                v2f b0 = *(const v2f*)(Wlds + m * W_STRIDE        + wOff + kk * 4 + bk);
                v2f b1 = *(const v2f*)(Wlds + (16 + m) * W_STRIDE + wOff + kk * 4 + bk);
                acc0 = __builtin_amdgcn_wmma_f32_16x16x4_f32(
                    false, a, false, b0, (short)0, acc0, false, false);
                acc1 = __builtin_amdgcn_wmma_f32_16x16x4_f32(
                    false, a, false, b1, (short)0, acc1, false, false);
            }
        }

        // ===== bias + leaky_relu, spill logits to LDS (C layout -> [edge][anchor]) =====
        #pragma unroll
        for (int r = 0; r < 8; ++r) {
            int row = r + hs * 8;
            float v0 = acc0[r] + bl0;
            v0 = v0 >= 0.0f ? v0 : NEG_SLOPE * v0;
            float v1 = acc1[r] + bl1;
            v1 = v1 >= 0.0f ? v1 : NEG_SLOPE * v1;
            prb[row * PR_STRIDE + m]      = v0;
            prb[row * PR_STRIDE + 16 + m] = v1;
        }

        // ===== softmax over 32 anchors: lane -> (edge m, anchor-half hs) =====
        {
            float ev[16];
            float mx = -3.402823466e+38f;
            #pragma unroll
            for (int j = 0; j < 16; ++j) {
                ev[j] = prb[m * PR_STRIDE + hs * 16 + j];
                mx = fmaxf(mx, ev[j]);
            }
            mx = fmaxf(mx, __shfl_xor(mx, 16, 32));
            float s = 0.0f;
            #pragma unroll
            for (int j = 0; j < 16; ++j) { ev[j] = __expf(ev[j] - mx); s += ev[j]; }
            s += __shfl_xor(s, 16, 32);
            float inv = 1.0f / s;
            #pragma unroll
            for (int j = 0; j < 16; ++j)
                prb[m * PR_STRIDE + hs * 16 + j] = ev[j] * inv;
        }

        // ===== prompt GEMM: probs(16x32) @ anchor(32x128), 8 N-tiles x 8 K-steps =====
        v2f afr[8];
        #pragma unroll
        for (int kk = 0; kk < 8; ++kk)
            afr[kk] = *(const v2f*)(prb + m * PR_STRIDE + kk * 4 + bk);

        #pragma unroll
        for (int t = 0; t < 8; ++t) {
            v8f acc = {};
            #pragma unroll
            for (int kk = 0; kk < 8; ++kk) {
                v2f b = *(const v2f*)(aT + (t * 16 + m) * AT_STRIDE + kk * 4 + bk);
                acc = __builtin_amdgcn_wmma_f32_16x16x4_f32(
                    false, afr[kk], false, b, (short)0, acc, false, false);
            }
            // stash C fragments into the (now idle) comb tile for coalesced stores
            #pragma unroll
            for (int r = 0; r < 8; ++r)
                comb[(r + hs * 8) * COMB_STRIDE + t * 16 + m] = acc[r];
        }

        // ===== coalesced b128 row stores: one 512B row per instruction =====
        #pragma unroll 4
        for (int i = 0; i < 16; ++i) {
            int e2 = e0 + i;
            if (e2 < E) {
                v4f val = *(const v4f*)(comb + i * COMB_STRIDE + lane * 4);
                *(v4f*)(out + (size_t)e2 * DIM + lane * 4) = val;
            }
        }
    }
}

extern "C" void kernel_launch(void* const* d_in, const int* in_sizes, int n_in,
                              void* d_out, int out_size, void* d_ws, size_t ws_size,
                              hipStream_t stream) {
    const float* x          = (const float*)d_in[0];
    const int*   edge_index = (const int*)d_in[1];
    // d_in[2] = layer (scalar, unused)
    const float* W          = (const float*)d_in[3];
    const float* b_lin      = (const float*)d_in[4];
    const float* anchor     = (const float*)d_in[5];
    float* out = (float*)d_out;

    const int E = in_sizes[1] / 2;
    const int nTiles = (E + 15) / 16;
    int blocks = (nTiles + WAVES - 1) / WAVES;
    if (blocks > 512) blocks = 512;
    if (blocks < 1) blocks = 1;

    const size_t smemBytes = (size_t)SMEM_FLOATS * sizeof(float);
    edgeprompt_wmma_kernel<<<blocks, TPB, smemBytes, stream>>>(
        x, edge_index, W, b_lin, anchor, out, E);
}